// CRD_21715354649108
// MI455X (gfx1250) — compile-verified
//
#include <hip/hip_runtime.h>
#include <math.h>
#include <stdint.h>

// Problem constants (from reference)
#define B_     64
#define D_     128
#define K_     16384          // last valid k index; count = K_+1
#define KP1    16385
#define NDATA  1000000
#define INV_T  14.285714285714286f   // 1/0.07
#define MOM_   0.5f

#define WAVES  4              // waves per block in the dot kernel
#define LROW   132            // padded LDS row stride (floats) -> conflict-free
                              // b64 reads: banks 4*m + 4c + koff all distinct

typedef float v2f __attribute__((ext_vector_type(2)));
typedef float v8f __attribute__((ext_vector_type(8)));
typedef float f4v __attribute__((ext_vector_type(4)));
typedef int   gv4i __attribute__((vector_size(16)));   // matches builtin param

// ---------------------------------------------------------------------------
// CDNA5 async global->LDS staging (ASYNCcnt-tracked). Guarded: falls back to
// direct per-lane gathers if the builtin is not declared by this toolchain.
// ---------------------------------------------------------------------------
#if __has_builtin(__builtin_amdgcn_global_load_async_to_lds_b128)
#define USE_ASYNC_LDS 1
#else
#define USE_ASYNC_LDS 0
#endif

#if USE_ASYNC_LDS
static __device__ __forceinline__ void async_copy16B(const float* g, const float* l) {
    // AS1/AS3 typed-vector pointers recovered via integer round-trip
    // (LDS generic address: low 32 bits == LDS byte offset per the aperture
    // mapping). Param type per clang diagnostic: AS1 'int vector_size(16) *'.
    __builtin_amdgcn_global_load_async_to_lds_b128(
        (__attribute__((address_space(1))) gv4i*)(uintptr_t)g,
        (__attribute__((address_space(3))) gv4i*)(uint32_t)(uintptr_t)l,
        0, 0);
}
#endif

static __device__ __forceinline__ void wait_async0() {
    asm volatile("s_wait_asynccnt 0x0" ::: "memory");
}

// ---------------------------------------------------------------------------
// Zero the two Z-sum accumulators in workspace (re-zeroed every call)
// ---------------------------------------------------------------------------
__global__ void CRD_zero_ws(float* ws) {
    if (threadIdx.x == 0) { ws[0] = 0.0f; ws[1] = 0.0f; }
}

// ---------------------------------------------------------------------------
// Gather + dot + exp via V_WMMA_F32_16X16X4_F32.
// One wave computes a 16-row k-tile for one batch element b:
//   stage: 16 async b128 ops move 16 gathered 512B bank rows into this wave's
//          LDS slice (one fully-coalesced 512B row per instruction)
//   math:  A = 16 rows x 4-d-chunk from LDS; B = v[b] chunk replicated into
//          all 16 columns (B's N index never enters the address);
//          32 WMMA accumulations cover D=128.
// Writes exp(dot/T) unnormalized and atomically accumulates the Z partial sum.
// ---------------------------------------------------------------------------
__global__ __launch_bounds__(WAVES * 32) void CRD_nce_dot_wmma(
    const float* __restrict__ w,      // (NDATA, D) memory bank
    const float* __restrict__ v,      // (B, D)
    const int*   __restrict__ idx,    // (B, K+1)
    float*       __restrict__ out,    // (B, K+1) unnormalized exp
    float*       __restrict__ sum_acc)// Z partial-sum accumulator
{
#if USE_ASYNC_LDS
    __shared__ float smem[WAVES * 16 * LROW];   // 4 waves * 8448B slices
#endif
    const int lane  = threadIdx.x & 31;
    const int wave  = threadIdx.x >> 5;
    const int tile  = blockIdx.x * WAVES + wave;   // grid covers B*1025 exactly
    const int b     = tile / 1025;
    const int kt    = tile - b * 1025;
    const int m     = lane & 15;                   // A row / D column index
    const int khalf = lane >> 4;                   // lane-half selects K pair
    const int koff  = khalf * 2;
    const float* __restrict__ vrow = v + b * D_;

#if USE_ASYNC_LDS
    float* slice = smem + wave * (16 * LROW);
    #pragma unroll
    for (int r = 0; r < 16; ++r) {
        int kk = kt * 16 + r;
        kk = (kk <= K_) ? kk : K_;                 // clamp partial last tile
        const int row = idx[b * KP1 + kk];         // wave-uniform -> scalar load
        async_copy16B(w + (size_t)row * D_ + lane * 4,   // 32 lanes x 16B = row
                      slice + r * LROW + lane * 4);
    }
    wait_async0();
    const float* arow = slice + m * LROW;          // lane's A row (LDS)
#else
    const int k   = kt * 16 + m;
    const int kc  = (k <= K_) ? k : K_;
    const int row = idx[b * KP1 + kc];
    const float* arow = w + (size_t)row * D_;
    __builtin_prefetch(arow + khalf * 64,      0, 3);
    __builtin_prefetch(arow + khalf * 64 + 32, 0, 3);
#endif

    v8f acc = {0.f, 0.f, 0.f, 0.f, 0.f, 0.f, 0.f, 0.f};
    #pragma unroll
    for (int c = 0; c < 32; ++c) {
        const int d = c * 4 + koff;
        // A: lane holds row m, K = {koff, koff+1} of this 4-chunk (ds_load_b64)
        v2f a  = *(const v2f*)(arow + d);
        // B: same K striping (vgpr fast, lane-half high) -> columns replicate v
        v2f bb = *(const v2f*)(vrow + d);
        acc = __builtin_amdgcn_wmma_f32_16x16x4_f32(
                  false, a, false, bb, (short)0, acc, false, false);
    }

    // D layout: lane l, vgpr r -> row (r + 8*khalf), column l%16 (all equal).
    if (m == 0) {
        float lsum = 0.0f;
        #pragma unroll
        for (int r = 0; r < 8; ++r) {
            const int mm = r + khalf * 8;
            const int kk = kt * 16 + mm;
            if (kk <= K_) {
                const float e = __expf(acc[r] * INV_T);
                out[b * KP1 + kk] = e;
                lsum += e;
            }
        }
        atomicAdd(sum_acc, lsum);
    }
}

// ---------------------------------------------------------------------------
// Divide by Z; out_v1 and out_v2 are contiguous, Z_i = mean * NDATA
// ---------------------------------------------------------------------------
__global__ __launch_bounds__(256) void CRD_nce_norm(
    float* __restrict__ out, const float* __restrict__ ws)
{
    const float cnt = (float)(B_ * KP1);
    const float r1  = 1.0f / (ws[0] * ((float)NDATA / cnt));
    const float r2  = 1.0f / (ws[1] * ((float)NDATA / cnt));
    const int   n   = B_ * KP1;
    const int   stride = gridDim.x * blockDim.x;
    for (int i = blockIdx.x * blockDim.x + threadIdx.x; i < 2 * n; i += stride)
        out[i] *= (i < n) ? r1 : r2;
}

// ---------------------------------------------------------------------------
// Streaming bank copy (512 MB each), non-temporal to spare L2 for the gathers
// ---------------------------------------------------------------------------
__global__ __launch_bounds__(256) void CRD_bank_copy(
    const f4v* __restrict__ src, f4v* __restrict__ dst, long n4)
{
    const long stride = (long)gridDim.x * blockDim.x;
    for (long i = (long)blockIdx.x * blockDim.x + threadIdx.x; i < n4; i += stride) {
        f4v t = __builtin_nontemporal_load(&src[i]);
        __builtin_nontemporal_store(t, &dst[i]);
    }
}

// ---------------------------------------------------------------------------
// Momentum update + l2 renorm of the 64 selected rows in each bank
// ---------------------------------------------------------------------------
__global__ void CRD_bank_update(
    const float* __restrict__ mem1, const float* __restrict__ v1,
    const float* __restrict__ mem2, const float* __restrict__ v2,
    const int*   __restrict__ y,
    float* __restrict__ nm1, float* __restrict__ nm2)
{
    const int t = blockIdx.x * blockDim.x + threadIdx.x;
    if (t >= 2 * B_) return;
    const int bank = t >> 6;
    const int b    = t & 63;
    const float* mem = bank ? mem2 : mem1;
    const float* v   = bank ? v2   : v1;
    float*       nm  = bank ? nm2  : nm1;
    const size_t ro  = (size_t)y[b] * D_;
    const size_t vo  = (size_t)b * D_;

    float ss = 0.0f;
    for (int d = 0; d < D_; ++d) {
        const float p = mem[ro + d] * MOM_ + v[vo + d] * (1.0f - MOM_);
        ss += p * p;
    }
    const float rn = rsqrtf(ss);
    for (int d = 0; d < D_; ++d) {
        const float p = mem[ro + d] * MOM_ + v[vo + d] * (1.0f - MOM_);
        nm[ro + d] = p * rn;
    }
}

// ---------------------------------------------------------------------------
extern "C" void kernel_launch(void* const* d_in, const int* in_sizes, int n_in,
                              void* d_out, int out_size, void* d_ws, size_t ws_size,
                              hipStream_t stream) {
    const float* v1     = (const float*)d_in[0];
    const float* v2     = (const float*)d_in[1];
    const int*   y      = (const int*)  d_in[2];
    const int*   idx    = (const int*)  d_in[3];
    const float* mem_v1 = (const float*)d_in[4];
    const float* mem_v2 = (const float*)d_in[5];

    float* out    = (float*)d_out;
    float* out_v1 = out;
    float* out_v2 = out + (size_t)B_ * KP1;
    float* nm1    = out + 2 * (size_t)B_ * KP1;
    float* nm2    = nm1 + (size_t)NDATA * D_;
    float* ws     = (float*)d_ws;

    CRD_zero_ws<<<1, 32, 0, stream>>>(ws);

    const int tiles  = B_ * 1025;          // 65600 16-row tiles (incl. partial)
    const int blocks = tiles / WAVES;      // exact cover

    // out_v1 = exp(<mem_v2[idx], v1>/T); out_v2 = exp(<mem_v1[idx], v2>/T)
    CRD_nce_dot_wmma<<<blocks, WAVES * 32, 0, stream>>>(mem_v2, v1, idx, out_v1, ws + 0);
    CRD_nce_dot_wmma<<<blocks, WAVES * 32, 0, stream>>>(mem_v1, v2, idx, out_v2, ws + 1);

    CRD_nce_norm<<<2048, 256, 0, stream>>>(out, ws);

    const long n4 = (long)NDATA * D_ / 4;  // 32M float4 per bank
    CRD_bank_copy<<<8192, 256, 0, stream>>>((const f4v*)mem_v1, (f4v*)nm1, n4);
    CRD_bank_copy<<<8192, 256, 0, stream>>>((const f4v*)mem_v2, (f4v*)nm2, n4);

    CRD_bank_update<<<1, 2 * B_, 0, stream>>>(mem_v1, v1, mem_v2, v2, y, nm1, nm2);
}